// SSIMWithLogits_82695300317755
// MI455X (gfx1250) — compile-verified
//
#include <hip/hip_runtime.h>

typedef __attribute__((ext_vector_type(2))) float v2f;
typedef __attribute__((ext_vector_type(8))) float v8f;

#define BATCH      16
#define HW         1024
#define TILES_X    (HW / 16)              // 64
#define TILES_IMG  (TILES_X * TILES_X)    // 4096
#define NTILES     (BATCH * TILES_IMG)    // 65536
#define WPB        4                      // waves (tiles) per block
#define NBLOCKS    (NTILES / WPB)         // 16384
#define NPIX       ((long long)BATCH * HW * HW)

#define PSTR       28                     // plane stride: 32 rows x 28 cols (K-padded)
#define PSZ        (32 * PSTR)            // 896 floats
#define HSTR       36                     // HT stride: 16 cols x 36 rows (bank-conflict-free, 16B-aligned)

// Normalized 1-D Gaussian, WS=11, sigma=1.5 (matches reference kernel)
__constant__ float GW[11] = {
    0.00102837f, 0.00759877f, 0.03600077f, 0.10936070f, 0.21300553f,
    0.26601172f,
    0.21300553f, 0.10936070f, 0.03600077f, 0.00759877f, 0.00102837f};

__device__ __forceinline__ int reflect_idx(int i) {
    if (i < 0) i = -i;                    // jnp 'reflect': -1 -> 1
    if (i >= HW) i = 2 * (HW - 1) - i;    // 1024 -> 1022
    return i;
}

// D = A(16x4 f32) * B(4x16 f32) + C on the CDNA5 fp32 matrix pipe
__device__ __forceinline__ v8f wmma4(v2f a, v2f b, v8f c) {
    return __builtin_amdgcn_wmma_f32_16x16x4_f32(
        false, a, false, b, (short)0, c, false, false);
}

__global__ void __launch_bounds__(WPB * 32)
ssim_tile_kernel(const float* __restrict__ inp, const float* __restrict__ tgt,
                 float* __restrict__ partials) {
    // Per-wave LDS slabs (52.2 KB / block, well under WGP capacity)
    __shared__ float s_l1[WPB][PSZ];        // sigmoid(input), padded 32x28, zeros outside 26x26
    __shared__ float s_l2[WPB][PSZ];        // target, padded 32x28
    __shared__ float s_p [WPB][PSZ];        // product plane, padded 32x28
    __shared__ float s_ht[WPB][16 * HSTR];  // H-pass result, column-major [n][k]
    __shared__ float s_part[WPB];

    const int lane = threadIdx.x & 31;
    const int wv   = threadIdx.x >> 5;
    const int tile = blockIdx.x * WPB + wv;
    const int b    = tile / TILES_IMG;
    const int rem  = tile - b * TILES_IMG;
    const int tr   = rem / TILES_X;
    const int tc   = rem - tr * TILES_X;
    const int y0   = tr * 16 - 5;           // halo origin
    const int x0   = tc * 16 - 5;

    const float* img1 = inp + (size_t)b * HW * HW;
    const float* img2 = tgt + (size_t)b * HW * HW;
    float* L1 = s_l1[wv];
    float* L2 = s_l2[wv];
    float* P  = s_p[wv];
    float* HT = s_ht[wv];

    // ---- Load 26x26 halo into padded 32x28 layout; reflect + sigmoid ----
    for (int t = lane; t < PSZ; t += 32) {
        int r = t / PSTR, c = t - r * PSTR;
        float v1 = 0.0f, v2 = 0.0f;
        if (r < 26 && c < 26) {
            int gy = reflect_idx(y0 + r);
            int gx = reflect_idx(x0 + c);
            float a = img1[gy * HW + gx];
            v2 = img2[gy * HW + gx];
            v1 = 1.0f / (1.0f + __expf(-a));   // sigmoid == exp(logsigmoid)
        }
        L1[t] = v1;
        L2[t] = v2;
    }

    // ---- Banded Gaussian weight fragments (shared by A- and B-side) ----
    // A 16x4 layout: lanes 0-15 = M; VGPR0 holds K+0 / K+2, VGPR1 K+1 / K+3.
    // B 4x16 assumed to mirror C/D half-stride striping, so W[k][i]=g[k-i]
    // built once serves both sides.
    const int half = lane >> 4;             // 0: lanes 0-15, 1: lanes 16-31
    const int mn   = lane & 15;             // M (A-side) == N (B-side)
    v2f Wf[7];
#pragma unroll
    for (int c = 0; c < 7; ++c) {
        int k0 = 4 * c + 2 * half;
        int j0 = k0 - mn, j1 = j0 + 1;
        Wf[c].x = (j0 >= 0 && j0 <= 10) ? GW[j0] : 0.0f;
        Wf[c].y = (j1 >= 0 && j1 <= 10) ? GW[j1] : 0.0f;
    }

    // Hoisted LDS addressing
    const int arow_t = mn * PSTR;           // A rows 0..15  (region rows)
    const int arow_b = (16 + mn) * PSTR;    // A rows 16..31 (26..31 are zeros)
    const int hcol   = mn * HSTR;           // HT column base
    const int row0   = 8 * half;            // D-layout: element r holds M = r + 8*half
    const int v4     = lane * 4;            // vectorized plane-build offset

    __syncthreads();

    // Separable 11-tap filter of one padded plane as banded GEMMs:
    //   H: htmp(32x16) = X(32x28) . Wh(28x16)   (two 16-row GEMMs, K=28)
    //   V: out(16x16)  = Wv(16x28) . htmp(28x16)
    auto conv2d = [&](const float* PL) -> v8f {
        v8f top = {}, bot = {};
#pragma unroll
        for (int c = 0; c < 7; ++c) {
            int k0 = 4 * c + 2 * half;
            v2f at = *(const v2f*)(PL + arow_t + k0);
            v2f ab = *(const v2f*)(PL + arow_b + k0);
            top = wmma4(at, Wf[c], top);
            bot = wmma4(ab, Wf[c], bot);
        }
        // Column-major staging: lane's 8 rows are contiguous -> b128 stores,
        // no conditionals (rows 28..31 are zeros, never read back).
        *(float4*)(HT + hcol + row0)          = make_float4(top[0], top[1], top[2], top[3]);
        *(float4*)(HT + hcol + row0 + 4)      = make_float4(top[4], top[5], top[6], top[7]);
        *(float4*)(HT + hcol + 16 + row0)     = make_float4(bot[0], bot[1], bot[2], bot[3]);
        *(float4*)(HT + hcol + 16 + row0 + 4) = make_float4(bot[4], bot[5], bot[6], bot[7]);
        __syncthreads();

        v8f o = {};
#pragma unroll
        for (int c = 0; c < 7; ++c) {
            int k0 = 4 * c + 2 * half;
            v2f bf = *(const v2f*)(HT + hcol + k0);   // single ds_load_b64
            o = wmma4(Wf[c], bf, o);
        }
        __syncthreads();        // HT / plane buffers reused next pass
        return o;
    };

    v8f acc[5];
    acc[0] = conv2d(L1);        // mu1
    acc[1] = conv2d(L2);        // mu2

    // Product planes: pure vectorized LDS streams (7 x b128 per buffer)
#pragma unroll
    for (int it = 0; it < 7; ++it) {
        int idx = it * 128 + v4;
        float4 a = *(const float4*)(L1 + idx);
        *(float4*)(P + idx) = make_float4(a.x * a.x, a.y * a.y, a.z * a.z, a.w * a.w);
    }
    __syncthreads();
    acc[2] = conv2d(P);         // E[x^2]

#pragma unroll
    for (int it = 0; it < 7; ++it) {
        int idx = it * 128 + v4;
        float4 a = *(const float4*)(L2 + idx);
        *(float4*)(P + idx) = make_float4(a.x * a.x, a.y * a.y, a.z * a.z, a.w * a.w);
    }
    __syncthreads();
    acc[3] = conv2d(P);         // E[y^2]

#pragma unroll
    for (int it = 0; it < 7; ++it) {
        int idx = it * 128 + v4;
        float4 a = *(const float4*)(L1 + idx);
        float4 bb = *(const float4*)(L2 + idx);
        *(float4*)(P + idx) = make_float4(a.x * bb.x, a.y * bb.y, a.z * bb.z, a.w * bb.w);
    }
    __syncthreads();
    acc[4] = conv2d(P);         // E[xy]

    // ---- Pointwise SSIM map + clip + local sum (8 pixels per lane) ----
    const float C1f = 1.0e-4f;  // (0.01*1)^2
    const float C2f = 9.0e-4f;  // (0.03*1)^2
    float lsum = 0.0f;
#pragma unroll
    for (int r = 0; r < 8; ++r) {
        float mu1 = acc[0][r], mu2 = acc[1][r];
        float e11 = acc[2][r], e22 = acc[3][r], e12 = acc[4][r];
        float mu1sq = mu1 * mu1, mu2sq = mu2 * mu2, mu12 = mu1 * mu2;
        float sg1 = e11 - mu1sq, sg2 = e22 - mu2sq, sg12 = e12 - mu12;
        float num = (2.0f * mu12 + C1f) * (2.0f * sg12 + C2f);
        float den = (mu1sq + mu2sq + C1f) * (sg1 + sg2 + C2f);
        float l = 1.0f - num / den;
        l = fminf(fmaxf(l, 0.0f), 1.0f) * 0.5f;
        lsum += l;
    }

    // ---- Wave reduce (wave32), deterministic block partial ----
#pragma unroll
    for (int off = 16; off > 0; off >>= 1)
        lsum += __shfl_down(lsum, off, 32);
    if (lane == 0) s_part[wv] = lsum;
    __syncthreads();
    if (threadIdx.x == 0) {
        float s = 0.0f;
#pragma unroll
        for (int w = 0; w < WPB; ++w) s += s_part[w];
        partials[blockIdx.x] = s;
    }
}

__global__ void __launch_bounds__(256)
ssim_reduce_kernel(const float* __restrict__ partials, float* __restrict__ out) {
    __shared__ float sm[256];
    float s = 0.0f;
    for (int i = threadIdx.x; i < NBLOCKS; i += 256) s += partials[i];
    sm[threadIdx.x] = s;
    __syncthreads();
    for (int st = 128; st > 0; st >>= 1) {
        if (threadIdx.x < st) sm[threadIdx.x] += sm[threadIdx.x + st];
        __syncthreads();
    }
    if (threadIdx.x == 0) out[0] = sm[0] * (float)(1.0 / (double)NPIX);
}

extern "C" void kernel_launch(void* const* d_in, const int* in_sizes, int n_in,
                              void* d_out, int out_size, void* d_ws, size_t ws_size,
                              hipStream_t stream) {
    const float* inp = (const float*)d_in[0];
    const float* tgt = (const float*)d_in[1];
    float* out = (float*)d_out;
    float* partials = (float*)d_ws;    // 16384 floats = 64 KB scratch

    ssim_tile_kernel<<<NBLOCKS, WPB * 32, 0, stream>>>(inp, tgt, partials);
    ssim_reduce_kernel<<<1, 256, 0, stream>>>(partials, out);
}